// embedding_concat_11879879541683
// MI455X (gfx1250) — compile-verified
//
#include <hip/hip_runtime.h>
#include <hip/hip_bf16.h>

// out[b, c, h, w] = (c < C1) ? x[b, c, h, w] : y[b, c-C1, h>>1, w>>1]
// Shapes: x (176, 64, 56, 56) f32, y (176, 128, 28, 28) f32,
//         out (176, 192, 56, 56) f32.
// Pure bandwidth problem (~636 MB traffic, ~27 us floor at 23.3 TB/s).
// Geometry carries all the structure: grid.y = batch (scalar SALU bases),
// blockDim = (14,16) = 7 wave32 so threadIdx.x == w4 and no runtime
// division or modulo survives; grid covers the output exactly (no tail).
// Each wave stores 32 consecutive float4s = one coalesced 512B NT B128 store.

typedef float v4f __attribute__((ext_vector_type(4)));
typedef float v2f __attribute__((ext_vector_type(2)));

namespace {
constexpr unsigned B   = 176;
constexpr unsigned C1  = 64;
constexpr unsigned C2  = 128;
constexpr unsigned H1  = 56;
constexpr unsigned W4  = 14;               // 56/4 float4s per output row
constexpr unsigned XB4 = C1 * H1 * W4;     // 50176 x float4s per batch
constexpr unsigned OB4 = (C1 + C2) * H1 * W4;  // 150528 out float4s per batch
constexpr unsigned Y2B = C2 * (H1 / 2) * W4;   // 50176 y float2s per batch
constexpr unsigned ROWS_PER_BLK = 16;      // threadIdx.y rows per block
constexpr unsigned XBLOCKS = (C1 * H1) / ROWS_PER_BLK;  // 224 (3584 x-rows)
constexpr unsigned YBLOCKS = (C2 * H1) / ROWS_PER_BLK;  // 448 (7168 out-rows)
}  // namespace

__global__ __launch_bounds__(224) void upcat_b128_nt_kernel(
    const float* __restrict__ x, const float* __restrict__ y,
    float* __restrict__ out) {
  const unsigned w4 = threadIdx.x;         // 0..13  (== float4 column)
  const unsigned ty = threadIdx.y;         // 0..15  (row within block)
  const unsigned b  = blockIdx.y;          // batch (scalar)

  const v4f* __restrict__ x4 = reinterpret_cast<const v4f*>(x);
  const v2f* __restrict__ y2 = reinterpret_cast<const v2f*>(y);
  v4f* __restrict__ out4 = reinterpret_cast<v4f*>(out);

  if (blockIdx.x < XBLOCKS) {
    // x half: straight copy. Flat per-batch float4 index, batch base scalar.
    const unsigned j = (blockIdx.x * ROWS_PER_BLK + ty) * W4 + w4;  // < 50176
    const v4f v = __builtin_nontemporal_load(x4 + (b * XB4 + j));
    __builtin_nontemporal_store(v, out4 + (b * OB4 + j));
  } else {
    // y half: r = c2*56 + h in [0,7168). 56 even => r>>1 = c2*28 + (h>>1),
    // so y float2 index = (r>>1)*14 + w4 (8-byte aligned, 2x reuse -> RT).
    const unsigned r = (blockIdx.x - XBLOCKS) * ROWS_PER_BLK + ty;
    const v2f u = y2[b * Y2B + (r >> 1) * W4 + w4];
    v4f v;
    v.x = u.x; v.y = u.x; v.z = u.y; v.w = u.y;
    __builtin_nontemporal_store(v, out4 + (b * OB4 + XB4 + r * W4 + w4));
  }
}

extern "C" void kernel_launch(void* const* d_in, const int* in_sizes, int n_in,
                              void* d_out, int out_size, void* d_ws, size_t ws_size,
                              hipStream_t stream) {
  (void)in_sizes; (void)n_in; (void)out_size; (void)d_ws; (void)ws_size;
  const float* x = reinterpret_cast<const float*>(d_in[0]);
  const float* y = reinterpret_cast<const float*>(d_in[1]);
  float* out = reinterpret_cast<float*>(d_out);

  const dim3 grid(XBLOCKS + YBLOCKS, B, 1);   // (672, 176): exact cover, no tail
  const dim3 block(W4, ROWS_PER_BLK, 1);      // (14, 16) = 224 = 7 wave32
  upcat_b128_nt_kernel<<<grid, block, 0, stream>>>(x, y, out);
}